// RelGATConvLayer_69389491634569
// MI455X (gfx1250) — compile-verified
//
#include <hip/hip_runtime.h>
#include <hip/hip_bf16.h>
#include <math.h>

// ---------------- problem constants (match reference) ----------------
#define N_NODES   100000
#define IN_FEAT   128
#define NUM_HEADS 4
#define OUT_FEAT  32
#define HID       128            // NUM_HEADS * OUT_FEAT
#define NUM_RELS  3
#define NUM_EDGES 400000
#define NEG_SLOPE 0.2f

#define ROW_TILES (N_NODES / 16) // 6250, exact

typedef float v2f __attribute__((ext_vector_type(2)));
typedef float v8f __attribute__((ext_vector_type(8)));

// XOR swizzle for the LDS-resident W^T tile: keeps allocation at exactly
// 128*128 floats (64 KB) while making the 16-lane B-fragment loads hit 16
// distinct banks. Preserves (k, k+1) adjacency and 8-byte alignment since it
// only touches bits [5:2] of the k index.
__device__ __forceinline__ int wswz(int n, int k) {
    return n * IN_FEAT + (k ^ ((n & 15) << 2));
}

// ---------------------------------------------------------------------
// h = x @ W_r  (100000x128 @ 128x128, fp32) via V_WMMA_F32_16X16X4_F32.
// Block = 256 threads = 8 waves; wave w owns row-tile (blockIdx.x*8 + w),
// i.e. a 16x128 slab of h, computed as 8 column tiles of 16x16.
// ---------------------------------------------------------------------
__global__ __launch_bounds__(256) void gemm_h_kernel(
    const float* __restrict__ x, const float* __restrict__ Wr,
    float* __restrict__ h) {
    __shared__ float ldsW[IN_FEAT * HID]; // 64 KB, W transposed + swizzled

    // Stage W_r (row-major [k][n]) into LDS as swizzled W^T.
    for (int i = threadIdx.x; i < IN_FEAT * HID; i += 256) {
        int k = i >> 7;      // input feature
        int n = i & 127;     // output feature
        ldsW[wswz(n, k)] = Wr[i];
    }
    __syncthreads();

    int waveId  = threadIdx.x >> 5;
    int lane    = threadIdx.x & 31;
    int rowTile = blockIdx.x * 8 + waveId;
    if (rowTile >= ROW_TILES) rowTile = ROW_TILES - 1; // keep EXEC all-ones

    int halfSel = lane >> 4;        // 0: lanes 0-15, 1: lanes 16-31
    int lid     = lane & 15;
    int koff    = halfSel << 1;     // A frag: K = k0 + {0,1} or {2,3}

    const float* xp = x + (size_t)(rowTile * 16 + lid) * IN_FEAT + koff;

    v8f acc[8] = {}; // 8 column tiles of 16x16 f32 accumulators

    #pragma unroll 4
    for (int k0 = 0; k0 < IN_FEAT; k0 += 4) {
        // A: 16x4 f32 fragment (2 VGPRs, 8B-aligned pair load)
        v2f a = *(const v2f*)(xp + k0);
        #pragma unroll
        for (int t = 0; t < 8; ++t) {
            // B: 4x16 f32 fragment from LDS W^T (lane holds col=lid, K pair)
            int n = t * 16 + lid;
            v2f b = *(const v2f*)(&ldsW[wswz(n, k0 + koff)]);
            acc[t] = __builtin_amdgcn_wmma_f32_16x16x4_f32(
                false, a, false, b, (short)0, acc[t], false, false);
        }
    }

    // C/D layout: VGPR j -> row j (+8 for lanes 16-31), col = lid.
    int rowBase = rowTile * 16 + (halfSel << 3);
    #pragma unroll
    for (int t = 0; t < 8; ++t) {
        #pragma unroll
        for (int j = 0; j < 8; ++j) {
            h[(size_t)(rowBase + j) * HID + t * 16 + lid] = acc[t][j];
        }
    }
}

// ---------------------------------------------------------------------
// el[n,h] = <h[n,h,:], attn_l[h,:]> ; er likewise.
// ---------------------------------------------------------------------
__global__ __launch_bounds__(256) void attn_dots_kernel(
    const float* __restrict__ h, const float* __restrict__ al,
    const float* __restrict__ ar, float* __restrict__ el,
    float* __restrict__ er) {
    int idx = blockIdx.x * 256 + threadIdx.x;
    if (idx >= N_NODES * NUM_HEADS) return;
    int node = idx >> 2, head = idx & 3;
    const float4* hp  = (const float4*)(h + (size_t)node * HID + head * OUT_FEAT);
    const float4* alp = (const float4*)(al + head * OUT_FEAT);
    const float4* arp = (const float4*)(ar + head * OUT_FEAT);
    float sl = 0.f, sr = 0.f;
    #pragma unroll
    for (int d = 0; d < OUT_FEAT / 4; ++d) {
        float4 hv = hp[d], av = alp[d], bv = arp[d];
        sl += hv.x * av.x + hv.y * av.y + hv.z * av.z + hv.w * av.w;
        sr += hv.x * bv.x + hv.y * bv.y + hv.z * bv.z + hv.w * bv.w;
    }
    el[idx] = sl;
    er[idx] = sr;
}

// Monotone encoding so unsigned-int max == float max.
__device__ __forceinline__ unsigned fenc(float f) {
    unsigned u = __float_as_uint(f);
    return (u & 0x80000000u) ? ~u : (u | 0x80000000u);
}
__device__ __forceinline__ float fdec(unsigned k) {
    return (k & 0x80000000u) ? __uint_as_float(k & 0x7FFFFFFFu)
                             : __uint_as_float(~k);
}

__global__ __launch_bounds__(256) void init_md_kernel(
    unsigned* __restrict__ m, float* __restrict__ denom) {
    int idx = blockIdx.x * 256 + threadIdx.x;
    if (idx >= N_NODES * NUM_HEADS) return;
    m[idx] = 0u;      // encoded identity (below every real key)
    denom[idx] = 0.f;
}

// e = leaky_relu(el[src] + er[dst]); segment max over dst via atomicMax(u32).
__global__ __launch_bounds__(256) void edge_score_max_kernel(
    const int* __restrict__ src, const int* __restrict__ dst,
    const float* __restrict__ el, const float* __restrict__ er,
    float* __restrict__ e_edge, unsigned* __restrict__ m) {
    int idx = blockIdx.x * 256 + threadIdx.x;
    if (idx >= NUM_EDGES * NUM_HEADS) return;
    int e = idx >> 2, head = idx & 3;
    int s = src[e], d = dst[e];
    float v = el[s * NUM_HEADS + head] + er[d * NUM_HEADS + head];
    v = (v > 0.f) ? v : NEG_SLOPE * v;
    e_edge[idx] = v;
    atomicMax(&m[d * NUM_HEADS + head], fenc(v));
}

// ex = exp(e - m[dst]); denom[dst] += ex (ex overwrites e_edge in place).
__global__ __launch_bounds__(256) void edge_exp_sum_kernel(
    const int* __restrict__ dst, float* __restrict__ e_edge,
    const unsigned* __restrict__ m, float* __restrict__ denom) {
    int idx = blockIdx.x * 256 + threadIdx.x;
    if (idx >= NUM_EDGES * NUM_HEADS) return;
    int e = idx >> 2, head = idx & 3;
    int d = dst[e];
    float ex = __expf(e_edge[idx] - fdec(m[d * NUM_HEADS + head]));
    e_edge[idx] = ex;
    atomicAdd(&denom[d * NUM_HEADS + head], ex);
}

// out[dst] += alpha * h[src]; one wave per edge, lane = feature within head.
__global__ __launch_bounds__(256) void edge_aggregate_kernel(
    const int* __restrict__ src, const int* __restrict__ dst,
    const float* __restrict__ ex, const float* __restrict__ denom,
    const float* __restrict__ h, float* __restrict__ out) {
    int gid  = blockIdx.x * 256 + threadIdx.x;
    int e    = gid >> 5;
    int lane = gid & 31;
    if (e >= NUM_EDGES) return;
    int s = src[e], d = dst[e];
    #pragma unroll
    for (int head = 0; head < NUM_HEADS; ++head) {
        float alpha = ex[e * NUM_HEADS + head] / denom[d * NUM_HEADS + head];
        float val   = alpha * h[(size_t)s * HID + head * OUT_FEAT + lane];
        atomicAdd(&out[(size_t)d * HID + head * OUT_FEAT + lane], val);
    }
}

// out[n, :] = sum_r bias[r, :]  (accumulation target for all relations)
__global__ __launch_bounds__(256) void init_out_kernel(
    const float* __restrict__ bias, float* __restrict__ out) {
    int idx = blockIdx.x * 256 + threadIdx.x;
    if (idx >= N_NODES * HID) return;
    int j = idx & (HID - 1);
    out[idx] = bias[j] + bias[HID + j] + bias[2 * HID + j];
}

// ---------------------------------------------------------------------
extern "C" void kernel_launch(void* const* d_in, const int* in_sizes, int n_in,
                              void* d_out, int out_size, void* d_ws,
                              size_t ws_size, hipStream_t stream) {
    const float* x      = (const float*)d_in[0]; // [N, 128]
    const float* W      = (const float*)d_in[1]; // [3, 128, 128]
    const float* attn_l = (const float*)d_in[2]; // [3, 4, 32]
    const float* attn_r = (const float*)d_in[3]; // [3, 4, 32]
    const float* bias   = (const float*)d_in[4]; // [3, 4, 32]
    const int*   eidx   = (const int*)d_in[5];   // [3, 2, E]
    float* out = (float*)d_out;                  // [N, 4, 32]

    // Workspace carve-up (relations processed sequentially, buffers reused):
    char* ws = (char*)d_ws;
    float*    h      = (float*)ws;    ws += (size_t)N_NODES * HID * sizeof(float);       // 51.2 MB
    float*    el     = (float*)ws;    ws += (size_t)N_NODES * NUM_HEADS * sizeof(float); // 1.6 MB
    float*    er     = (float*)ws;    ws += (size_t)N_NODES * NUM_HEADS * sizeof(float);
    unsigned* m      = (unsigned*)ws; ws += (size_t)N_NODES * NUM_HEADS * sizeof(unsigned);
    float*    denom  = (float*)ws;    ws += (size_t)N_NODES * NUM_HEADS * sizeof(float);
    float*    e_edge = (float*)ws;    // E * 4 floats (scores, then exp in place)

    init_out_kernel<<<(N_NODES * HID + 255) / 256, 256, 0, stream>>>(bias, out);

    for (int r = 0; r < NUM_RELS; ++r) {
        const float* Wr  = W + (size_t)r * IN_FEAT * HID;
        const float* alr = attn_l + (size_t)r * NUM_HEADS * OUT_FEAT;
        const float* arr = attn_r + (size_t)r * NUM_HEADS * OUT_FEAT;
        const int*   src = eidx + (size_t)r * 2 * NUM_EDGES;
        const int*   dst = src + NUM_EDGES;

        gemm_h_kernel<<<(ROW_TILES + 7) / 8, 256, 0, stream>>>(x, Wr, h);
        attn_dots_kernel<<<(N_NODES * NUM_HEADS + 255) / 256, 256, 0, stream>>>(
            h, alr, arr, el, er);
        init_md_kernel<<<(N_NODES * NUM_HEADS + 255) / 256, 256, 0, stream>>>(
            m, denom);
        edge_score_max_kernel<<<(NUM_EDGES * NUM_HEADS + 255) / 256, 256, 0,
                                stream>>>(src, dst, el, er, e_edge, m);
        edge_exp_sum_kernel<<<(NUM_EDGES * NUM_HEADS + 255) / 256, 256, 0,
                              stream>>>(dst, e_edge, m, denom);
        edge_aggregate_kernel<<<(NUM_EDGES * 32 + 255) / 256, 256, 0, stream>>>(
            src, dst, e_edge, denom, h, out);
    }
}